// RnnClassifier_56547539419195
// MI455X (gfx1250) — compile-verified
//
#include <hip/hip_runtime.h>
#include <math.h>

// Problem constants (from reference)
#define V_  50000
#define E_  128
#define H_  15
#define G_  45      // 3*H
#define GP_ 48      // padded gate dim
#define C_  20
#define B_  512
#define S_  512

typedef __attribute__((ext_vector_type(2))) float v2f;
typedef __attribute__((ext_vector_type(8))) float v8f;

// ---------------------------------------------------------------------------
// Kernel 1: fused embedding gather + gates_x GEMM via V_WMMA_F32_16X16X4_F32.
//   gates[s][g][b] = sum_e embed[x[b,s]][e] * W_ih[g][e] + b_ih[g]
// One wave per 16-row tile (fixed s, 16 consecutive b). 3 N-tiles accumulated
// simultaneously so gathered A fragments are reused 3x.
// ---------------------------------------------------------------------------
__global__ void __launch_bounds__(256)
gates_wmma_kernel(const int* __restrict__ x,
                  const float* __restrict__ embed,
                  const float* __restrict__ W_ih,
                  const float* __restrict__ b_ih,
                  float* __restrict__ gates)
{
    const int lane = threadIdx.x & 31;
    const int wave = blockIdx.x * (blockDim.x >> 5) + (threadIdx.x >> 5); // rowtile
    const int s      = wave >> 5;          // B/16 = 32 rowtiles per timestep
    const int b_base = (wave & 31) << 4;   // first batch row of this tile
    const int half   = lane >> 4;          // K-half select (0: K=0,1  1: K=2,3)
    const int lm     = lane & 15;          // M (for A) / N (for B,C,D)

    // Token for matrix row M = lm  (x is [B,S] row-major)
    const int token = x[(b_base + lm) * S_ + s];
    const float* arow = embed + token * E_ + 2 * half;

    // B fragment source rows: B[k][n] = W_ih[nb + n][k]
    const int gn0 = lm, gn1 = 16 + lm, gn2 = 32 + lm;
    const bool v2 = (gn2 < G_);
    const float* brow0 = W_ih + gn0 * E_ + 2 * half;
    const float* brow1 = W_ih + gn1 * E_ + 2 * half;
    const float* brow2 = W_ih + (v2 ? gn2 : 0) * E_ + 2 * half; // clamp: no OOB,
                                                                // bad cols never stored
    v8f c0 = {}, c1 = {}, c2 = {};
    #pragma unroll
    for (int k0 = 0; k0 < E_; k0 += 4) {
        v2f a  = *(const v2f*)(arow  + k0);
        v2f b0 = *(const v2f*)(brow0 + k0);
        v2f b1 = *(const v2f*)(brow1 + k0);
        v2f b2 = *(const v2f*)(brow2 + k0);
        c0 = __builtin_amdgcn_wmma_f32_16x16x4_f32(false, a, false, b0,
                                                   (short)0, c0, false, false);
        c1 = __builtin_amdgcn_wmma_f32_16x16x4_f32(false, a, false, b1,
                                                   (short)0, c1, false, false);
        c2 = __builtin_amdgcn_wmma_f32_16x16x4_f32(false, a, false, b2,
                                                   (short)0, c2, false, false);
    }

    // Fuse b_ih and store. D layout: lane lm = N, VGPR v -> M = v + 8*half.
    const float bias0 = b_ih[gn0];
    const float bias1 = b_ih[gn1];
    const float bias2 = b_ih[v2 ? gn2 : 0];

    // gates layout [S][GP_][B]: addr = s*GP_*B + g*B + b,  b = b_base + M
    const int base = s * (GP_ * B_) + b_base + 8 * half;
    #pragma unroll
    for (int v = 0; v < 8; ++v) {
        gates[base + gn0 * B_ + v] = c0[v] + bias0;
        gates[base + gn1 * B_ + v] = c1[v] + bias1;
        if (v2) gates[base + gn2 * B_ + v] = c2[v] + bias2;
    }
}

// ---------------------------------------------------------------------------
// Kernel 2: sequential GRU scan + output head + softmax.
// One thread per batch element; h lives in VGPRs; W_hh/b_hh/W_out/b_out are
// wave-uniform -> scalar-cache loads co-issued with VALU FMAs. gates reads
// are lane-coalesced thanks to the [S,48,B] layout.
// ---------------------------------------------------------------------------
__global__ void __launch_bounds__(32)
gru_scan_kernel(const float* __restrict__ gates,
                const float* __restrict__ W_hh,
                const float* __restrict__ b_hh,
                const float* __restrict__ W_out,
                const float* __restrict__ b_out,
                float* __restrict__ out)
{
    const int b = blockIdx.x * blockDim.x + threadIdx.x;   // 0..511

    float h[H_];
    #pragma unroll
    for (int j = 0; j < H_; ++j) h[j] = 0.0f;

    for (int s = 0; s < S_; ++s) {
        const float* gx = gates + s * (GP_ * B_) + b;
        float hnew[H_];
        #pragma unroll
        for (int j = 0; j < H_; ++j) {
            float hr = b_hh[j];
            float hz = b_hh[H_ + j];
            float hn = b_hh[2 * H_ + j];
            #pragma unroll
            for (int k = 0; k < H_; ++k) {
                const float hk = h[k];
                hr = fmaf(W_hh[j * H_ + k],            hk, hr);
                hz = fmaf(W_hh[(H_ + j) * H_ + k],     hk, hz);
                hn = fmaf(W_hh[(2 * H_ + j) * H_ + k], hk, hn);
            }
            const float xr = gx[j * B_];
            const float xz = gx[(H_ + j) * B_];
            const float xn = gx[(2 * H_ + j) * B_];
            const float r = 1.0f / (1.0f + __expf(-(xr + hr)));
            const float z = 1.0f / (1.0f + __expf(-(xz + hz)));
            const float n = tanhf(xn + r * hn);
            hnew[j] = (1.0f - z) * n + z * h[j];
        }
        #pragma unroll
        for (int j = 0; j < H_; ++j) h[j] = hnew[j];
    }

    // logits = h @ W_out.T + b_out ; softmax
    float logits[C_];
    float m = -3.0e38f;
    #pragma unroll
    for (int c = 0; c < C_; ++c) {
        float l = b_out[c];
        #pragma unroll
        for (int j = 0; j < H_; ++j) l = fmaf(W_out[c * H_ + j], h[j], l);
        logits[c] = l;
        m = fmaxf(m, l);
    }
    float sum = 0.0f;
    #pragma unroll
    for (int c = 0; c < C_; ++c) {
        logits[c] = __expf(logits[c] - m);
        sum += logits[c];
    }
    const float inv = 1.0f / sum;
    #pragma unroll
    for (int c = 0; c < C_; ++c) out[b * C_ + c] = logits[c] * inv;
}

// ---------------------------------------------------------------------------
extern "C" void kernel_launch(void* const* d_in, const int* in_sizes, int n_in,
                              void* d_out, int out_size, void* d_ws, size_t ws_size,
                              hipStream_t stream)
{
    const int*   x     = (const int*)  d_in[0];
    const float* embed = (const float*)d_in[1];
    const float* W_ih  = (const float*)d_in[2];
    const float* b_ih  = (const float*)d_in[3];
    const float* W_hh  = (const float*)d_in[4];
    const float* b_hh  = (const float*)d_in[5];
    const float* W_out = (const float*)d_in[6];
    const float* b_out = (const float*)d_in[7];
    float* out   = (float*)d_out;
    float* gates = (float*)d_ws;   // [S][48][B] fp32 = 50.3 MB scratch

    // (S*B/16) row tiles, 8 waves per 256-thread block
    const int rowtiles = (S_ * B_) / 16;            // 16384
    gates_wmma_kernel<<<rowtiles / 8, 256, 0, stream>>>(x, embed, W_ih, b_ih, gates);

    // 512 batch elements, 1 wave per block spread over 16 WGPs
    gru_scan_kernel<<<B_ / 32, 32, 0, stream>>>(gates, W_hh, b_hh, W_out, b_out, out);
}